// MultiHeadAttention_58488864637255
// MI455X (gfx1250) — compile-verified
//
#include <hip/hip_runtime.h>
#include <math.h>

// MI455X / gfx1250 MHA forward, fp32 end-to-end using V_WMMA_F32_16X16X4_F32.
// 45 GFLOP total vs ~6us HBM floor -> fp32 matrix path is fast enough and
// numerically safe vs the fp32 reference. Round 2: split the attention score
// accumulation into two independent WMMA chains (ILP across XDL latency) and
// prefetch the next K/V tile across the softmax serialization point.

typedef float v2f __attribute__((ext_vector_type(2)));
typedef float v8f __attribute__((ext_vector_type(8)));

#define EMBED   768
#define THREE_E 2304
#define NHEAD   12
#define HDIM    64
#define BATCH   2
#define SEQ     2048

__device__ __forceinline__ v8f wmma4(v2f a, v2f b, v8f c) {
    // D = A(16x4,f32) * B(4x16,f32) + C(16x16,f32)
    return __builtin_amdgcn_wmma_f32_16x16x4_f32(false, a, false, b,
                                                 (short)0, c, false, false);
}

// C[M,N] = A[M,K] @ B[K,N] + bias[N]; one wave computes a 64x64 tile (4x4 WMMA frags).
__global__ __launch_bounds__(128) void gemm_bias_f32_wmma(
    const float* __restrict__ A, const float* __restrict__ B,
    const float* __restrict__ bias, float* __restrict__ C,
    int M, int N, int K)
{
    const int wave = (int)((blockIdx.x * blockDim.x + threadIdx.x) >> 5);
    const int lane = (int)(threadIdx.x & 31u);
    const int nt   = N >> 6;                 // 64-wide N tiles
    const int tm   = wave / nt;
    const int tn   = wave - tm * nt;
    if (tm * 64 >= M) return;                // wave-uniform guard

    const int r    = lane & 15;
    const int hi   = lane >> 4;
    const int koff = hi * 2;                 // lane<16 -> K {0,1}, lane>=16 -> K {2,3}

    v8f zero = {};
    v8f acc[4][4];
#pragma unroll
    for (int mi = 0; mi < 4; ++mi)
#pragma unroll
        for (int ni = 0; ni < 4; ++ni) acc[mi][ni] = zero;

    const float* Arow[4];
#pragma unroll
    for (int mi = 0; mi < 4; ++mi)
        Arow[mi] = A + (size_t)(tm * 64 + mi * 16 + r) * K + koff;
    const float* Bcol[4];
#pragma unroll
    for (int ni = 0; ni < 4; ++ni)
        Bcol[ni] = B + (size_t)koff * N + tn * 64 + ni * 16 + r;

    for (int k = 0; k < K; k += 4) {
        v2f af[4];
#pragma unroll
        for (int mi = 0; mi < 4; ++mi) {
            const float* ap = Arow[mi] + k;
            v2f a = {ap[0], ap[1]};
            af[mi] = a;
        }
        v2f bf[4];
#pragma unroll
        for (int ni = 0; ni < 4; ++ni) {
            const float* bp = Bcol[ni] + (size_t)k * N;
            v2f b = {bp[0], bp[N]};
            bf[ni] = b;
        }
#pragma unroll
        for (int mi = 0; mi < 4; ++mi)
#pragma unroll
            for (int ni = 0; ni < 4; ++ni)
                acc[mi][ni] = wmma4(af[mi], bf[ni], acc[mi][ni]);
    }

#pragma unroll
    for (int mi = 0; mi < 4; ++mi)
#pragma unroll
        for (int ni = 0; ni < 4; ++ni) {
            const int col = tn * 64 + ni * 16 + r;
            const float bv = bias[col];
#pragma unroll
            for (int i = 0; i < 8; ++i) {
                const int row = tm * 64 + mi * 16 + i + 8 * hi;
                C[(size_t)row * N + col] = acc[mi][ni][i] + bv;
            }
        }
}

// Flash-attention: one block per (b, h, 64-query tile); 4 waves, 16 query rows each.
// qkv layout: [B, S, 3, H, D] row stride THREE_E. out: [B, S, E].
#define LROW 17  // padded LDS row stride (bank spread)

__global__ __launch_bounds__(128) void attn_wmma(const float* __restrict__ qkv,
                                                 float* __restrict__ out)
{
    __shared__ float ls[4][16 * LROW];  // per-wave 16x16 score/prob tile
    __shared__ float lm[4][16];         // running max per row
    __shared__ float ll[4][16];         // running sum per row
    __shared__ float lc[4][16];         // correction factor per row

    const int w    = (int)(threadIdx.x >> 5);
    const int lane = (int)(threadIdx.x & 31u);
    const int r    = lane & 15;
    const int hi   = lane >> 4;
    const int koff = hi * 2;

    const int blk  = (int)blockIdx.x;
    const int bIdx = blk / (NHEAD * (SEQ / 64));
    const int rem  = blk % (NHEAD * (SEQ / 64));
    const int h    = rem / (SEQ / 64);
    const int qt   = rem % (SEQ / 64);
    const int qbase = qt * 64 + w * 16;   // this wave's 16 query rows

    const float* base = qkv + (size_t)bIdx * SEQ * THREE_E;
    const int qoff  = h * HDIM;
    const int koffh = EMBED + h * HDIM;
    const int voffh = 2 * EMBED + h * HDIM;

    // Preload Q fragments: 16 chunks of 16x4 covering D=64
    v2f qf[16];
    {
        const float* qrow = base + (size_t)(qbase + r) * THREE_E + qoff + koff;
#pragma unroll
        for (int kc = 0; kc < 16; ++kc) {
            v2f a = {qrow[4 * kc], qrow[4 * kc + 1]};
            qf[kc] = a;
        }
    }

    if (lane < 16) { lm[w][lane] = -INFINITY; ll[w][lane] = 0.0f; }
    __syncthreads();

    v8f zero = {};
    v8f o[4];
#pragma unroll
    for (int dt = 0; dt < 4; ++dt) o[dt] = zero;

    for (int kt = 0; kt < SEQ / 16; ++kt) {
        // ---- prefetch next tile's K and V rows (2 x 128B halves per 256B row,
        // upper lane half selects the half-line) while this tile computes
        if (kt + 1 < SEQ / 16) {
            const size_t nrow = (size_t)((kt + 1) * 16 + r) * THREE_E;
            __builtin_prefetch(base + nrow + koffh + hi * 32, 0, 3);
            __builtin_prefetch(base + nrow + voffh + hi * 32, 0, 3);
        }

        // ---- scores: S(16x16) = Q(16x64) @ K^T(64x16), scaled by 1/sqrt(64)
        // Two independent accumulation chains to cover XDL latency.
        v8f sc0 = zero, sc1 = zero;
        const float* krow = base + (size_t)(kt * 16 + r) * THREE_E + koffh + koff;
#pragma unroll
        for (int kc = 0; kc < 8; ++kc) {
            v2f b0 = {krow[4 * kc], krow[4 * kc + 1]};
            sc0 = wmma4(qf[kc], b0, sc0);
            v2f b1 = {krow[4 * (kc + 8)], krow[4 * (kc + 8) + 1]};
            sc1 = wmma4(qf[kc + 8], b1, sc1);
        }
#pragma unroll
        for (int i = 0; i < 8; ++i)
            ls[w][(i + 8 * hi) * LROW + r] = (sc0[i] + sc1[i]) * 0.125f;
        __syncthreads();

        // ---- online softmax bookkeeping (one lane per query row)
        if (lane < 16) {
            float* row = &ls[w][lane * LROW];
            float mOld = lm[w][lane];
            float tmax = -INFINITY;
#pragma unroll
            for (int c = 0; c < 16; ++c) tmax = fmaxf(tmax, row[c]);
            float mNew = fmaxf(mOld, tmax);
            float corr = __expf(mOld - mNew);
            float s = 0.0f;
#pragma unroll
            for (int c = 0; c < 16; ++c) {
                float p = __expf(row[c] - mNew);
                row[c] = p;
                s += p;
            }
            lm[w][lane] = mNew;
            ll[w][lane] = ll[w][lane] * corr + s;
            lc[w][lane] = corr;
        }
        __syncthreads();

        // ---- rescale O accumulators by per-row correction
        float cr[8];
#pragma unroll
        for (int i = 0; i < 8; ++i) cr[i] = lc[w][i + 8 * hi];
#pragma unroll
        for (int dt = 0; dt < 4; ++dt)
#pragma unroll
            for (int i = 0; i < 8; ++i) o[dt][i] = o[dt][i] * cr[i];

        // ---- O += P(16x16) @ V(16x64): 4 independent chains (one per d-tile)
        v2f pf[4];
#pragma unroll
        for (int kc = 0; kc < 4; ++kc) {
            v2f a = {ls[w][r * LROW + 4 * kc + koff],
                     ls[w][r * LROW + 4 * kc + koff + 1]};
            pf[kc] = a;
        }
#pragma unroll
        for (int dt = 0; dt < 4; ++dt) {
#pragma unroll
            for (int kc = 0; kc < 4; ++kc) {
                const float* vp = base + (size_t)(kt * 16 + kc * 4 + koff) * THREE_E
                                  + voffh + dt * 16 + r;
                v2f b = {vp[0], vp[THREE_E]};
                o[dt] = wmma4(pf[kc], b, o[dt]);
            }
        }
        __syncthreads();
    }

    // ---- normalize and store to attention output [B, S, E]
    float inv[8];
#pragma unroll
    for (int i = 0; i < 8; ++i) inv[i] = 1.0f / ll[w][i + 8 * hi];
#pragma unroll
    for (int dt = 0; dt < 4; ++dt)
#pragma unroll
        for (int i = 0; i < 8; ++i) {
            const size_t srow = (size_t)bIdx * SEQ + qbase + i + 8 * hi;
            out[srow * EMBED + h * HDIM + dt * 16 + r] = o[dt][i] * inv[i];
        }
}

extern "C" void kernel_launch(void* const* d_in, const int* in_sizes, int n_in,
                              void* d_out, int out_size, void* d_ws, size_t ws_size,
                              hipStream_t stream) {
    (void)in_sizes; (void)n_in; (void)out_size; (void)ws_size;
    const float* x      = (const float*)d_in[0];
    const float* w_qkv  = (const float*)d_in[1];
    const float* b_qkv  = (const float*)d_in[2];
    const float* w_proj = (const float*)d_in[3];
    const float* b_proj = (const float*)d_in[4];
    float* out = (float*)d_out;

    float* qkv  = (float*)d_ws;                                  // [B*S, 3E] 37.7 MB
    float* attn = qkv + (size_t)BATCH * SEQ * THREE_E;           // [B*S, E]  12.6 MB

    const int M = BATCH * SEQ;  // 4096

    // QKV projection: 64x64 tile per wave, 4 waves per block
    {
        int waves = (M / 64) * (THREE_E / 64);   // 64*36 = 2304
        gemm_bias_f32_wmma<<<waves / 4, 128, 0, stream>>>(x, w_qkv, b_qkv, qkv,
                                                          M, THREE_E, EMBED);
    }
    // Flash attention
    {
        int blocks = BATCH * NHEAD * (SEQ / 64); // 768
        attn_wmma<<<blocks, 128, 0, stream>>>(qkv, attn);
    }
    // Output projection
    {
        int waves = (M / 64) * (EMBED / 64);     // 64*12 = 768
        gemm_bias_f32_wmma<<<waves / 4, 128, 0, stream>>>(attn, w_proj, b_proj, out,
                                                          M, EMBED, EMBED);
    }
}